// SE3HamNODEGT_28338194219311
// MI455X (gfx1250) — compile-verified
//
#include <hip/hip_runtime.h>
#include <hip/hip_bf16.h>
#include <stdint.h>

// ---------------------------------------------------------------------------
// SE3 Hamiltonian NODE ground-truth dynamics, MI455X (gfx1250).
// Memory-bound streaming op (~0.57 FLOP/byte, ~15us HBM roofline): staged
// through LDS with the CDNA5 Tensor Data Mover (tensor_load_to_lds /
// tensor_store_from_lds), synchronized with s_wait_tensorcnt.
// One row (22 f32) per thread.
// Toolchain: clang-23 lane -> 6-arg TDM builtins
//   (uint32x4 g0, int32x8 g1, int32x4 g2, int32x4 g3, int32x8 extra, i32 cpol)
// ---------------------------------------------------------------------------

typedef uint32_t u32;
typedef uint64_t u64;
typedef __attribute__((ext_vector_type(4))) u32 v4u32;
typedef __attribute__((ext_vector_type(8))) int v8i32;
typedef __attribute__((ext_vector_type(4))) int v4i32;

#define FEAT 22
#define ROWS 128                       // rows per block (one wave32 issues TDM)
#define TILE_FLOATS (ROWS * FEAT)      // 2816 elements
#define TILE_BYTES  (TILE_FLOATS * 4)  // 11264 bytes

// ---- per-row physics (constants baked from the reference) ------------------
__device__ __forceinline__ void se3_row(const float* __restrict__ x,
                                        float* __restrict__ y) {
  const float c_m1   = 0.027f;                       // M1 diag  (inv(_M*I))
  const float c_minv = 37.037037037037035f;          // M1_INV diag (_M)
  const float jx     = 2.3951e-05f;                  // J diag xy
  const float jz     = 3.2347e-05f;                  // J diag z
  const float jix    = (float)(1.0 / 2.3951e-05);    // J^-1 diag xy
  const float jiz    = (float)(1.0 / 3.2347e-05);    // J^-1 diag z
  const float mg     = (float)(0.027 * 9.81);        // m*g

  const float r00 = x[3],  r01 = x[4],  r02 = x[5];
  const float r10 = x[6],  r11 = x[7],  r12 = x[8];
  const float r20 = x[9],  r21 = x[10], r22 = x[11];
  const float qvx = x[12], qvy = x[13], qvz = x[14];
  const float qwx = x[15], qwy = x[16], qwz = x[17];
  const float u0  = x[18], u1 = x[19], u2 = x[20], u3 = x[21];

  // momenta
  const float pvx = qvx * c_m1, pvy = qvy * c_m1, pvz = qvz * c_m1;
  const float pwx = qwx * jx,   pwy = qwy * jx,   pwz = qwz * jz;
  // dH/dp
  const float hvx = pvx * c_minv, hvy = pvy * c_minv, hvz = pvz * c_minv;
  const float hwx = pwx * jix,    hwy = pwy * jix,    hwz = pwz * jiz;

  // dx = R @ dHdpv
  y[0] = r00 * hvx + r01 * hvy + r02 * hvz;
  y[1] = r10 * hvx + r11 * hvy + r12 * hvz;
  y[2] = r20 * hvx + r21 * hvy + r22 * hvz;

  // dR rows: cross(R_i, dHdpw)
  y[3]  = r01 * hwz - r02 * hwy;
  y[4]  = r02 * hwx - r00 * hwz;
  y[5]  = r00 * hwy - r01 * hwx;
  y[6]  = r11 * hwz - r12 * hwy;
  y[7]  = r12 * hwx - r10 * hwz;
  y[8]  = r10 * hwy - r11 * hwx;
  y[9]  = r21 * hwz - r22 * hwy;
  y[10] = r22 * hwx - r20 * hwz;
  y[11] = r20 * hwy - r21 * hwx;

  // dpv = cross(pv, hw) - mg*R_2 + [0,0,u0]
  const float dpvx = (pvy * hwz - pvz * hwy) - mg * r20;
  const float dpvy = (pvz * hwx - pvx * hwz) - mg * r21;
  const float dpvz = (pvx * hwy - pvy * hwx) - mg * r22 + u0;

  // dpw = cross(pw, hw) + cross(pv, hv) + [u1,u2,u3]
  const float dpwx = (pwy * hwz - pwz * hwy) + (pvy * hvz - pvz * hvy) + u1;
  const float dpwy = (pwz * hwx - pwx * hwz) + (pvz * hvx - pvx * hvz) + u2;
  const float dpwz = (pwx * hwy - pwy * hwx) + (pvx * hvy - pvy * hvx) + u3;

  // dv, dw
  y[12] = dpvx * c_minv;
  y[13] = dpvy * c_minv;
  y[14] = dpvz * c_minv;
  y[15] = dpwx * jix;
  y[16] = dpwy * jix;
  y[17] = dpwz * jiz;
  // zeros_like(u)
  y[18] = 0.0f; y[19] = 0.0f; y[20] = 0.0f; y[21] = 0.0f;
}

// ---- TDM descriptor build: one 1-D tile of `nelem` f32 at gaddr <-> lds ----
__device__ __forceinline__ void tdm_build(u64 gaddr, u32 lds_off, u32 nelem,
                                          v4u32& g0, v8i32& g1,
                                          v4i32& g2, v4i32& g3) {
  // Group 0: count=1 (valid), lds_addr, 57-bit global_addr, type=2 ("image")
  g0.x = 1u;
  g0.y = lds_off;
  g0.z = (u32)gaddr;
  g0.w = ((u32)(gaddr >> 32) & 0x01FFFFFFu) | (2u << 30);

  // Group 1: data_size=2 (4 B), tensor_dim0 = tile_dim0 = nelem (each tile is
  // its own fully in-bounds tensor), tensor_dim1 = 1, tile_dim1/2 = 1,
  // stride0 = nelem.
  g1[0] = 0x00020000;                                   // data_size = 4B
  g1[1] = (int)((nelem & 0xFFFFu) << 16);               // tensor_dim0 lo16
  g1[2] = (int)((nelem >> 16) | (1u << 16));            // td0 hi16 | td1 lo16=1
  g1[3] = (int)((nelem & 0xFFFFu) << 16);               // tile_dim0 (16-bit)
  g1[4] = 0x00010001;                                   // tile_dim1=1, tile_dim2=1
  g1[5] = (int)nelem;                                   // tensor_dim0_stride lo
  g1[6] = 0;
  g1[7] = 0;

  // Groups 2/3: benign 5-D completion (dims=1, tiles degenerate).
  g2[0] = 1;            // tensor_dim2 = 1
  g2[1] = 1;            // tensor_dim3 = 1
  g2[2] = 0;            // tensor_dim2_stride lo
  g2[3] = 0x00010000;   // tile_dim3 = 1
  g3[0] = 0;            // tensor_dim3_stride lo
  g3[1] = 0x00010000;   // tensor_dim4 = 1 (lo16 in [31:16])
  g3[2] = 0x00010000;   // tile_dim4 = 1 ([31:16])
  g3[3] = 0;
}

__global__ __launch_bounds__(ROWS)
void se3_tile_kernel(const float* __restrict__ in, float* __restrict__ out) {
  __shared__ float s_in[TILE_FLOATS];
  __shared__ float s_out[TILE_FLOATS];

  const u32 tid = threadIdx.x;
  const u64 tile_byte = (u64)blockIdx.x * (u64)TILE_BYTES;

  // Wave 0: async DMA the whole input tile (global -> LDS) via the TDM.
  if (tid < 32u) {
    v4u32 g0; v8i32 g1; v4i32 g2, g3;
    tdm_build((u64)(uintptr_t)in + tile_byte,
              (u32)(uintptr_t)&s_in[0], TILE_FLOATS, g0, g1, g2, g3);
    const v8i32 xtra = {0, 0, 0, 0, 0, 0, 0, 0};
    __builtin_amdgcn_tensor_load_to_lds(g0, g1, g2, g3, xtra, 0);
    __builtin_amdgcn_s_wait_tensorcnt(0);
  }
  __syncthreads();

  // One row per thread, straight out of LDS.
  float yreg[FEAT];
  se3_row(&s_in[tid * FEAT], yreg);
#pragma unroll
  for (int k = 0; k < FEAT; ++k) s_out[tid * FEAT + k] = yreg[k];
  __syncthreads();

  // Wave 0: async DMA the result tile (LDS -> global) via the TDM.
  if (tid < 32u) {
    v4u32 g0; v8i32 g1; v4i32 g2, g3;
    tdm_build((u64)(uintptr_t)out + tile_byte,
              (u32)(uintptr_t)&s_out[0], TILE_FLOATS, g0, g1, g2, g3);
    const v8i32 xtra = {0, 0, 0, 0, 0, 0, 0, 0};
    __builtin_amdgcn_tensor_store_from_lds(g0, g1, g2, g3, xtra, 0);
    __builtin_amdgcn_s_wait_tensorcnt(0);
  }
}

// Tail path (nrows % ROWS != 0) — plain per-row global loads/stores.
__global__ void se3_tail_kernel(const float* __restrict__ in,
                                float* __restrict__ out,
                                int start_row, int nrows) {
  const int r = start_row + (int)(blockIdx.x * blockDim.x + threadIdx.x);
  if (r >= nrows) return;
  float x[FEAT], y[FEAT];
#pragma unroll
  for (int k = 0; k < FEAT; ++k) x[k] = in[(size_t)r * FEAT + k];
  se3_row(x, y);
#pragma unroll
  for (int k = 0; k < FEAT; ++k) out[(size_t)r * FEAT + k] = y[k];
}

extern "C" void kernel_launch(void* const* d_in, const int* in_sizes, int n_in,
                              void* d_out, int out_size, void* d_ws, size_t ws_size,
                              hipStream_t stream) {
  // d_in[0] = t (scalar, unused); d_in[1] = input (nrows x 22 f32)
  const float* x = (const float*)d_in[1];
  float* y = (float*)d_out;
  const int nrows  = in_sizes[1] / FEAT;
  const int ntiles = nrows / ROWS;

  if (ntiles > 0) {
    se3_tile_kernel<<<ntiles, ROWS, 0, stream>>>(x, y);
  }
  const int rem = nrows - ntiles * ROWS;
  if (rem > 0) {
    se3_tail_kernel<<<(rem + 127) / 128, 128, 0, stream>>>(x, y, ntiles * ROWS,
                                                           nrows);
  }
}